// PlaneElement__80857054314543
// MI455X (gfx1250) — compile-verified
//
#include <hip/hip_runtime.h>

typedef float v2f __attribute__((ext_vector_type(2)));
typedef float v8f __attribute__((ext_vector_type(8)));

#define EPSV 1e-9f

// ---------------------------------------------------------------------------
// Stage 1: grid-strided sum reduction of `area` using V_WMMA_F32_16X16X4_F32
// with an all-ones B matrix (D = A*1 + C accumulates row sums; summation is
// permutation-invariant, so arbitrary 64-float chunks serve as "A").
// Chunk = 512 floats per wave: 4 coalesced b128 loads in flight -> 8 WMMAs.
// One partial per block -> d_ws (no atomics: deterministic).
// ---------------------------------------------------------------------------
__global__ void __launch_bounds__(256)
reduce_wmma_kernel(const float* __restrict__ area,
                   float* __restrict__ partials,
                   long long nChunks) {
    const int lane          = threadIdx.x & 31;
    const int wv            = threadIdx.x >> 5;
    const int wavesPerBlock = blockDim.x >> 5;
    const long long waveId   = (long long)blockIdx.x * wavesPerBlock + wv;
    const long long numWaves = (long long)gridDim.x * wavesPerBlock;

    v8f acc = {0.f, 0.f, 0.f, 0.f, 0.f, 0.f, 0.f, 0.f};
    v2f ones; ones[0] = 1.0f; ones[1] = 1.0f;

    const float4* a4 = (const float4*)area;
    for (long long c = waveId; c < nChunks; c += numWaves) {
        // chunk c covers float4 indices [c*128, c*128+128) -> 512 floats.
        const float4* p = a4 + c * 128 + lane;
        float4 v0 = p[0];      // 4 outstanding b128 loads per wave
        float4 v1 = p[32];
        float4 v2 = p[64];
        float4 v3 = p[96];

        v2f x0; x0[0] = v0.x; x0[1] = v0.y;
        v2f x1; x1[0] = v0.z; x1[1] = v0.w;
        v2f x2; x2[0] = v1.x; x2[1] = v1.y;
        v2f x3; x3[0] = v1.z; x3[1] = v1.w;
        v2f x4; x4[0] = v2.x; x4[1] = v2.y;
        v2f x5; x5[0] = v2.z; x5[1] = v2.w;
        v2f x6; x6[0] = v3.x; x6[1] = v3.y;
        v2f x7; x7[0] = v3.z; x7[1] = v3.w;

        acc = __builtin_amdgcn_wmma_f32_16x16x4_f32(false, x0, false, ones, (short)0, acc, false, false);
        acc = __builtin_amdgcn_wmma_f32_16x16x4_f32(false, x1, false, ones, (short)0, acc, false, false);
        acc = __builtin_amdgcn_wmma_f32_16x16x4_f32(false, x2, false, ones, (short)0, acc, false, false);
        acc = __builtin_amdgcn_wmma_f32_16x16x4_f32(false, x3, false, ones, (short)0, acc, false, false);
        acc = __builtin_amdgcn_wmma_f32_16x16x4_f32(false, x4, false, ones, (short)0, acc, false, false);
        acc = __builtin_amdgcn_wmma_f32_16x16x4_f32(false, x5, false, ones, (short)0, acc, false, false);
        acc = __builtin_amdgcn_wmma_f32_16x16x4_f32(false, x6, false, ones, (short)0, acc, false, false);
        acc = __builtin_amdgcn_wmma_f32_16x16x4_f32(false, x7, false, ones, (short)0, acc, false, false);
    }

    // Sum this lane's 8 D elements, then cross-lane reduce (wave32).
    float local = acc[0] + acc[1] + acc[2] + acc[3] +
                  acc[4] + acc[5] + acc[6] + acc[7];
    for (int off = 16; off > 0; off >>= 1)
        local += __shfl_xor(local, off, 32);
    // Each row-sum is replicated across the 16 D columns -> divide by 16.
    __shared__ float wsum[8];
    if (lane == 0) wsum[wv] = local * (1.0f / 16.0f);
    __syncthreads();
    if (threadIdx.x == 0) {
        float s = 0.f;
        for (int i = 0; i < wavesPerBlock; ++i) s += wsum[i];
        partials[blockIdx.x] = s;
    }
}

// ---------------------------------------------------------------------------
// Stage 2: single block. Deterministically sum block partials + tail
// (< 512 elems), then compute the three scalar outputs. Only A[N-3..N-1]
// influence outflow_q: A_face[N-1] has zero slope (dA_p appended = 0), so
// only the last two faces and q_lat (via the global mean) matter.
// ---------------------------------------------------------------------------
__global__ void __launch_bounds__(256)
finalize_kernel(const float* __restrict__ area,
                const float* __restrict__ partials, int P,
                long long n, long long tailStart,
                const float* th_c, const float* F_cum, const float* rain,
                const float* dt_p, const float* WIDp, const float* MANp,
                const float* SLp, const float* dxp, const float* Ksp,
                const float* psip, const float* thsp,
                float* __restrict__ out) {
    __shared__ float sh[256];
    float s = 0.f;
    for (int i = threadIdx.x; i < P; i += 256) s += partials[i];
    for (long long i = tailStart + threadIdx.x; i < n; i += 256) s += area[i];
    sh[threadIdx.x] = s;
    __syncthreads();
    for (int stride = 128; stride > 0; stride >>= 1) {
        if (threadIdx.x < stride) sh[threadIdx.x] += sh[threadIdx.x + stride];
        __syncthreads();
    }
    if (threadIdx.x == 0) {
        float total = sh[0];

        const float WID = WIDp[0], MAN = MANp[0], SL = SLp[0], dx = dxp[0];
        const float dt  = dt_p[0];

        // Green-Ampt infiltration (scalar state, mean depth as head)
        float surface_head = (total / (float)n) / WID;
        float dtheta   = fmaxf(thsp[0] - th_c[0], 0.f);
        float f_cap    = Ksp[0] * (1.f + (psip[0] + surface_head) * dtheta /
                                         fmaxf(F_cum[0], EPSV));
        float supply   = rain[0] + surface_head / fmaxf(dt, EPSV);
        float infil_rate  = fmaxf(fminf(supply, f_cap), 0.f);
        float infil_depth = infil_rate * dt;

        float net_rain = fmaxf(rain[0] - infil_rate, 0.f);
        float q_lat    = net_rain * WID;

        // Manning: Q = A^(5/3) * sqrt(S) / (n * W^(2/3))
        const float coef = sqrtf(SL) / (MAN * powf(WID, 2.0f / 3.0f));
        const float A1 = area[n - 1], A2 = area[n - 2], A3 = area[n - 3];

        // Face at N-1: dA_p = 0 -> minmod = 0 -> A_face = max(A1, 0)
        float Qf_last = powf(fmaxf(A1, 0.f), 5.0f / 3.0f) * coef;
        // Face at N-2: full minmod slope
        float dm = A2 - A3, dp = A1 - A2;
        float mm = (dm * dp > 0.f) ? copysignf(fminf(fabsf(dm), fabsf(dp)), dm)
                                   : 0.f;
        float Af2     = fmaxf(A2 + 0.5f * mm, 0.f);
        float Qf_prev = powf(Af2, 5.0f / 3.0f) * coef;

        float A_next  = fmaxf(A1 + dt * (q_lat - (Qf_last - Qf_prev) / dx), 0.f);
        float outflow = powf(A_next, 5.0f / 3.0f) * coef;

        out[0] = outflow;
        out[1] = infil_rate;
        out[2] = infil_depth;
    }
}

extern "C" void kernel_launch(void* const* d_in, const int* in_sizes, int n_in,
                              void* d_out, int out_size, void* d_ws, size_t ws_size,
                              hipStream_t stream) {
    (void)n_in; (void)out_size; (void)ws_size;
    const float* area = (const float*)d_in[0];
    const long long n = (long long)in_sizes[0];

    float* partials = (float*)d_ws;        // P floats of scratch
    const int P = 512;
    const long long nChunks   = n / 512;   // 512 floats per wave-chunk
    const long long tailStart = nChunks * 512;

    reduce_wmma_kernel<<<P, 256, 0, stream>>>(area, partials, nChunks);

    finalize_kernel<<<1, 256, 0, stream>>>(
        area, partials, P, n, tailStart,
        (const float*)d_in[1],   // theta_current
        (const float*)d_in[2],   // F_cumulative
        (const float*)d_in[3],   // rain_rate
        (const float*)d_in[4],   // dt_s
        (const float*)d_in[6],   // WID
        (const float*)d_in[7],   // MAN
        (const float*)d_in[8],   // SL
        (const float*)d_in[9],   // dx
        (const float*)d_in[10],  // Ks
        (const float*)d_in[11],  // psi
        (const float*)d_in[12],  // theta_s
        (float*)d_out);
}